// DECOMP_59141699665997
// MI455X (gfx1250) — compile-verified
//
#include <hip/hip_runtime.h>

#define ASG __attribute__((address_space(1)))
#define ASL __attribute__((address_space(3)))

typedef const ASG char* gptr;
typedef ASL char*       lptr;

// 16-byte int vector matching the builtin's expected pointee type
// ('int __attribute__((__vector_size__(4 * sizeof(int))))').
typedef int v4i __attribute__((vector_size(16)));

namespace {
constexpr int Bn  = 128;
constexpr int Tn  = 768;
constexpr int Cn  = 256;
constexpr int TOT = Bn * Tn * Cn;           // 25,165,824 elements per output tensor
constexpr int TT  = 16;                     // timesteps staged per chunk
constexpr int NCH = Tn / TT;                // 48 chunks
constexpr int ROWB   = Cn * 4;              // 1024 B: global row stride == mirrored LDS row stride
constexpr int CHUNKB = TT * ROWB;           // 16 KiB LDS footprint per buffer
}

// ---- CDNA5 async global -> LDS copy (ASYNCcnt-tracked) -------------------
// OFF is a template parameter: the builtin requires a constant-integer offset.
template <int OFF>
__device__ __forceinline__ void async_ld_b128(gptr g, lptr l) {
#if __has_builtin(__builtin_amdgcn_global_load_async_to_lds_b128)
  __builtin_amdgcn_global_load_async_to_lds_b128(
      (ASG v4i*)g, (ASL v4i*)l, OFF, 0);
#else
  asm volatile("global_load_async_to_lds_b128 %0, %1, off offset:%2"
               :
               : "v"((unsigned)(unsigned long long)l), "v"(g), "i"(OFF)
               : "memory");
#endif
}

__device__ __forceinline__ void wait_async_16() {
#if __has_builtin(__builtin_amdgcn_s_wait_asynccnt)
  __builtin_amdgcn_s_wait_asynccnt(16);
#else
  asm volatile("s_wait_asynccnt 16" ::: "memory");
#endif
}

__device__ __forceinline__ void wait_async_0() {
#if __has_builtin(__builtin_amdgcn_s_wait_asynccnt)
  __builtin_amdgcn_s_wait_asynccnt(0);
#else
  asm volatile("s_wait_asynccnt 0" ::: "memory");
#endif
}

// One chunk = TT rows; lane L of wave 0 carries base+L*16; the immediate
// offset r*ROWB advances BOTH the global and LDS address (ISA async-load
// semantics), which matches the mirrored LDS layout exactly.
__device__ __forceinline__ void issue_chunk(gptr g, lptr l) {
  async_ld_b128< 0 * ROWB>(g, l);
  async_ld_b128< 1 * ROWB>(g, l);
  async_ld_b128< 2 * ROWB>(g, l);
  async_ld_b128< 3 * ROWB>(g, l);
  async_ld_b128< 4 * ROWB>(g, l);
  async_ld_b128< 5 * ROWB>(g, l);
  async_ld_b128< 6 * ROWB>(g, l);
  async_ld_b128< 7 * ROWB>(g, l);
  async_ld_b128< 8 * ROWB>(g, l);
  async_ld_b128< 9 * ROWB>(g, l);
  async_ld_b128<10 * ROWB>(g, l);
  async_ld_b128<11 * ROWB>(g, l);
  async_ld_b128<12 * ROWB>(g, l);
  async_ld_b128<13 * ROWB>(g, l);
  async_ld_b128<14 * ROWB>(g, l);
  async_ld_b128<15 * ROWB>(g, l);
}

__global__ __launch_bounds__(128) void DECOMP_59141699665997_kernel(
    const float* __restrict__ x, const float* __restrict__ ap,
    const float* __restrict__ bp, float* __restrict__ out) {
  __shared__ char smem[2 * CHUNKB];          // 32 KiB double buffer

  const int tid   = threadIdx.x;             // 0..127 -> one channel each
  const int blk   = blockIdx.x;              // 0..255
  const int bb    = blk >> 1;                // batch
  const int chalf = blk & 1;                 // which half of C

  const float alpha = *ap;
  const float beta  = *bp;
  const float ia = 1.0f - alpha;
  const float ib = 1.0f - beta;

  // element index of (bb, t=0, chalf*128)
  const int base_e = (bb * Tn) * Cn + chalf * 128;

  gptr g0 = (gptr)(unsigned long long)(const void*)x + (long long)base_e * 4;
  lptr l0 = (lptr)(unsigned)(unsigned long long)(void*)smem;

  gptr glane = g0 + tid * 16;                // only meaningful for wave 0
  lptr llane = l0 + tid * 16;

  // Prologue: wave 0 stages chunk 0 -> buf0 and chunk 1 -> buf1.
  if (tid < 32) {
    issue_chunk(glane, llane);
    issue_chunk(glane + CHUNKB, llane + CHUNKB);
  }
  gptr gnext = glane + 2 * CHUNKB;

  float s  = 0.0f;
  float bt = 0.0f;
  float* resp = out + base_e + tid;          // res stream
  float* map  = out + TOT + base_e + tid;    // ma  stream

  // ---------------- chunk 0 (peeled: t = 0, 1 initialization) ------------
  wait_async_16();                           // chunk 0 landed (chunk 1 in flight)
  __syncthreads();
  {
    lptr lb = l0;                            // buf0
    float x0 = *(const ASL float*)(lb + 0 * ROWB + tid * 4);
    s = x0;                                  // s0 = x[:,0,:]
    __builtin_nontemporal_store(x0 - s, resp);   // exactly 0
    __builtin_nontemporal_store(s, map);
    resp += Cn; map += Cn;

    float x1 = *(const ASL float*)(lb + 1 * ROWB + tid * 4);
    bt = x1 - s;                             // b0 = x[:,1,:] - s0
    {
      float sp = s;
      s  = fmaf(alpha, x1, ia * (sp + bt));
      bt = fmaf(beta, s - sp, ib * bt);
    }
    __builtin_nontemporal_store(x1 - s, resp);
    __builtin_nontemporal_store(s, map);
    resp += Cn; map += Cn;

#pragma unroll
    for (int r = 2; r < TT; ++r) {
      float xt = *(const ASL float*)(lb + r * ROWB + tid * 4);
      float sp = s;
      s  = fmaf(alpha, xt, ia * (sp + bt));
      bt = fmaf(beta, s - sp, ib * bt);
      __builtin_nontemporal_store(xt - s, resp);
      __builtin_nontemporal_store(s, map);
      resp += Cn; map += Cn;
    }
  }
  __syncthreads();                           // buf0 free for reuse
  if (tid < 32) {                            // chunk 2 -> buf0 (NCH = 48 > 2)
    issue_chunk(gnext, llane);
    gnext += CHUNKB;
  }

  // ---------------- steady-state chunks ----------------------------------
  for (int k = 1; k < NCH; ++k) {
    if (k < NCH - 1) wait_async_16();        // oldest outstanding chunk landed
    else             wait_async_0();
    __syncthreads();

    lptr lb = l0 + (k & 1) * CHUNKB;
#pragma unroll
    for (int r = 0; r < TT; ++r) {
      float xt = *(const ASL float*)(lb + r * ROWB + tid * 4);
      float sp = s;
      s  = fmaf(alpha, xt, ia * (sp + bt));
      bt = fmaf(beta, s - sp, ib * bt);
      __builtin_nontemporal_store(xt - s, resp);
      __builtin_nontemporal_store(s, map);
      resp += Cn; map += Cn;
    }

    __syncthreads();                         // buf[k&1] free
    if (tid < 32 && (k + 2) < NCH) {         // chunk k+2 -> buf[(k+2)&1] == buf[k&1]
      issue_chunk(gnext, llane + (k & 1) * CHUNKB);
      gnext += CHUNKB;
    }
  }
}

extern "C" void kernel_launch(void* const* d_in, const int* in_sizes, int n_in,
                              void* d_out, int out_size, void* d_ws, size_t ws_size,
                              hipStream_t stream) {
  (void)in_sizes; (void)n_in; (void)out_size; (void)d_ws; (void)ws_size;
  const float* x  = (const float*)d_in[0];
  const float* ap = (const float*)d_in[1];
  const float* bp = (const float*)d_in[2];
  float* out = (float*)d_out;

  dim3 grid(Bn * 2);    // 256 blocks: (batch, C-half)
  dim3 block(128);      // 4 waves; wave 0 drives the async pipeline
  DECOMP_59141699665997_kernel<<<grid, block, 0, stream>>>(x, ap, bp, out);
}